// Embedding_5583457485418
// MI455X (gfx1250) — compile-verified
//
#include <hip/hip_runtime.h>

// Embedding gather: out[row, :] = e[tok[row], :], row in [0, 8192), D = 1024 f32.
// Pure data movement (~64 MB HBM traffic => ~2.7 us floor at 23.3 TB/s; e fits
// entirely in the 192 MB L2, so steady-state replays are store-bound).
// CDNA5 async global<->LDS copies (ASYNCcnt-tracked): data never touches VGPRs.
// The async ops' INST_OFFSET advances BOTH the LDS and global addresses
// (CDNA5 ISA), so each direction needs only one address
// register set; the 8 x 512B chunks are addressed via offset: immediates.

#define EMB_DIM        1024              // floats per row
#define ROW_BYTES      4096              // EMB_DIM * 4
#define WAVES_PER_BLK  8                 // 256 threads / wave32
#define CHUNKS         8                 // 32 lanes * 16B * 8 = 4096B per row

__global__ __launch_bounds__(256) void embed_gather_async(
    const int* __restrict__ tok,
    const float* __restrict__ e,
    float* __restrict__ out,
    int n_rows)
{
    // Dynamic LDS: 8 waves * 4096B staging. No static __shared__ vars, so the
    // dynamic segment starts at LDS byte offset 0 and the wave-relative byte
    // offsets below are the literal LDS addresses the async ops consume.
    extern __shared__ char lds_stage[];
    (void)lds_stage;

    const int lane = threadIdx.x & 31;
    const int wave = threadIdx.x >> 5;
    int row = blockIdx.x * WAVES_PER_BLK + wave;
    const int row_stride = gridDim.x * WAVES_PER_BLK;

    const unsigned lds_base = (unsigned)(wave * ROW_BYTES + lane * 16);

    for (; row < n_rows; row += row_stride) {
        const int t = tok[row];  // wave-uniform
        const char* src = (const char*)(e   + (size_t)t   * EMB_DIM) + lane * 16;
        char*       dst = (char*)      (out + (size_t)row * EMB_DIM) + lane * 16;

        // Make sure last iteration's async stores drained before reusing LDS.
        asm volatile("s_wait_asynccnt 0" ::: "memory");

        // 8 x B128 async loads: global -> LDS. One LDS addr VGPR + one 64-bit
        // global addr pair; chunk stepping via the shared INST_OFFSET.
#pragma unroll
        for (int c = 0; c < CHUNKS; ++c) {
            asm volatile("global_load_async_to_lds_b128 %0, %1, off offset:%c2"
                         :: "v"(lds_base), "v"(src), "i"(c * 512)
                         : "memory");
        }

        // Wait for the row to land in LDS.
        asm volatile("s_wait_asynccnt 0" ::: "memory");

        // 8 x B128 async stores: LDS -> global.
#pragma unroll
        for (int c = 0; c < CHUNKS; ++c) {
            asm volatile("global_store_async_from_lds_b128 %0, %1, off offset:%c2"
                         :: "v"(dst), "v"(lds_base), "i"(c * 512)
                         : "memory");
        }
    }

    // Drain before wave exit (s_endpgm also implies wait-idle; explicit is safe).
    asm volatile("s_wait_asynccnt 0" ::: "memory");
}

extern "C" void kernel_launch(void* const* d_in, const int* in_sizes, int n_in,
                              void* d_out, int out_size, void* d_ws, size_t ws_size,
                              hipStream_t stream) {
    const int*   tok = (const int*)d_in[0];    // token_ids, 2*4096 = 8192 int32
    const float* e   = (const float*)d_in[1];  // [32000, 1024] f32
    float*       out = (float*)d_out;          // [8192, 1024] f32

    const int n_rows = in_sizes[0];            // 8192 token rows
    int blocks = (n_rows + WAVES_PER_BLK - 1) / WAVES_PER_BLK;  // 1024
    if (blocks > 4096) blocks = 4096;          // grid-stride covers the rest

    embed_gather_async<<<blocks, 256, WAVES_PER_BLK * ROW_BYTES, stream>>>(
        tok, e, out, n_rows);
}